// KVCache_35716948033553
// MI455X (gfx1250) — compile-verified
//
#include <hip/hip_runtime.h>
#include <stdint.h>

// KV-cache scatter insert for MI455X (gfx1250).
// Memory-bound: ~544 MiB total traffic -> ~24 us at 23.3 TB/s HBM.
// Bulk cache copy uses the CDNA5 async global<->LDS DMA path (ASYNCcnt),
// scatter uses b128 nontemporal loads/stores. No FLOPs -> no WMMA.

typedef __attribute__((ext_vector_type(4))) float v4f;

#define COPY_DEPTH 4

__device__ __forceinline__ void async_load_b128(uint32_t lds_addr, const void* gaddr) {
  // copies 16 bytes from the global address in gaddr to the LDS address in lds_addr (per lane, tracked by ASYNCcnt)
  asm volatile("global_load_async_to_lds_b128 %0, %1, off th:TH_LOAD_NT"
               :: "v"(lds_addr), "v"(gaddr)
               : "memory");
}

__device__ __forceinline__ void async_store_b128(const void* gaddr, uint32_t lds_addr) {
  // copies 16 bytes from the LDS address in lds_addr to the global address in gaddr (per lane, tracked by ASYNCcnt)
  asm volatile("global_store_async_from_lds_b128 %0, %1, off th:TH_STORE_NT"
               :: "v"(gaddr), "v"(lds_addr)
               : "memory");
}

__device__ __forceinline__ void wait_async0() {
  asm volatile("s_wait_asynccnt 0x0" ::: "memory");
}

// ---------------------------------------------------------------------------
// Pass 1: stream both caches into d_out through LDS with async DMA copies.
// One "unit" = 512 bytes = 32 lanes x b128. Each wave pipelines COPY_DEPTH
// units: 4 async loads -> wait -> 4 async stores -> wait. 8 waves/block hide
// the wait bubbles; a block iteration moves 16 KiB contiguous.
// ---------------------------------------------------------------------------
__global__ void __launch_bounds__(256)
kv_bulk_copy_async(const v4f* __restrict__ kc, const v4f* __restrict__ vc,
                   v4f* __restrict__ out, long long nF4, long long units)
{
  __shared__ v4f smem[256 * COPY_DEPTH];   // 16 KiB: 8 waves x 4 slots x 512 B
  const int lane = threadIdx.x & 31;       // wave32
  const int wv   = (int)(threadIdx.x >> 5);
  const int wpb  = (int)(blockDim.x >> 5);

  // Low 32 bits of a generic pointer to __shared__ == LDS byte offset.
  const uint32_t ldsWave = (uint32_t)(uintptr_t)(&smem[0])
                         + (uint32_t)(wv * (COPY_DEPTH * 512) + lane * 16);

  const long long stride = (long long)gridDim.x * wpb * COPY_DEPTH;
  for (long long u0 = ((long long)blockIdx.x * wpb + wv) * COPY_DEPTH;
       u0 < units; u0 += stride) {
#pragma unroll
    for (int j = 0; j < COPY_DEPTH; ++j) {
      long long u = u0 + j;                // wave-uniform -> EXEC stays full
      if (u < units) {
        long long f4 = u * 32 + lane;
        const v4f* src = (f4 < nF4) ? (kc + f4) : (vc + (f4 - nF4));
        async_load_b128(ldsWave + (uint32_t)(j * 512), (const void*)src);
      }
    }
    wait_async0();                         // loads landed in LDS
#pragma unroll
    for (int j = 0; j < COPY_DEPTH; ++j) {
      long long u = u0 + j;
      if (u < units) {
        long long f4 = u * 32 + lane;
        async_store_b128((const void*)(out + f4), ldsWave + (uint32_t)(j * 512));
      }
    }
    wait_async0();                         // buffers reusable next iteration
  }
}

// ---------------------------------------------------------------------------
// Pass 2: scatter k/v rows into the freshly written output.
// One thread = one b128 chunk of a 512 B row; rows land at pos[row] in the
// context axis, so every store burst is 512 B contiguous.
// ---------------------------------------------------------------------------
__global__ void __launch_bounds__(256)
kv_scatter(const v4f* __restrict__ k, const v4f* __restrict__ v,
           const int* __restrict__ pos, v4f* __restrict__ out,
           int chunk, int max_ctx, long long cacheF4, int nkv, int f4row)
{
  const long long perTensor = (long long)nkv * chunk * f4row;
  long long tid = (long long)blockIdx.x * blockDim.x + threadIdx.x;
  if (tid >= 2 * perTensor) return;

  const int  tensor = (int)(tid / perTensor);
  long long  r      = tid - (long long)tensor * perTensor;
  const int  d4     = (int)(r % f4row);
  long long  rr     = r / f4row;
  const int  row    = (int)(rr % chunk);
  const int  h      = (int)(rr / chunk);

  const int ctx = pos[row];

  const v4f* src = (tensor ? v : k)
                 + ((long long)h * chunk + row) * f4row + d4;
  v4f val = __builtin_nontemporal_load(src);

  v4f* dst = out + (long long)tensor * cacheF4
                 + ((long long)h * max_ctx + ctx) * f4row + d4;
  __builtin_nontemporal_store(val, dst);
}

extern "C" void kernel_launch(void* const* d_in, const int* in_sizes, int n_in,
                              void* d_out, int out_size, void* d_ws, size_t ws_size,
                              hipStream_t stream) {
  (void)d_ws; (void)ws_size; (void)out_size; (void)n_in;

  const float* kc  = (const float*)d_in[0];   // k_cache  (1, NKV, MAX_CTX, 128)
  const float* vc  = (const float*)d_in[1];   // v_cache
  const int*   pos = (const int*)  d_in[2];   // pos_ids  (CHUNK)
  const float* k   = (const float*)d_in[3];   // k        (1, NKV, CHUNK, 128)
  const float* v   = (const float*)d_in[4];   // v
  float* out = (float*)d_out;                 // [kout | vout] flat

  const int HEAD_DIM = 128;
  const long long nCacheF = in_sizes[0];                      // 33,554,432
  const int chunk = in_sizes[2];                              // 2048
  const long long nKF = in_sizes[3];                          // 2,097,152
  const int nkv = (int)(nKF / ((long long)chunk * HEAD_DIM)); // 8
  const int max_ctx = (int)(nCacheF / ((long long)nkv * HEAD_DIM)); // 32768

  // ---- Pass 1: bulk copy of both caches (256 MiB write) --------------------
  const long long nF4   = nCacheF / 4;          // float4s per cache tensor
  const long long units = (2 * nF4) / 32;       // 512 B units (exact division:
                                                // nCacheF is a multiple of 128)
  int copyBlocks = 2048;
  {
    long long needed = (units + (8LL * COPY_DEPTH) - 1) / (8LL * COPY_DEPTH);
    if (needed < copyBlocks) copyBlocks = (int)(needed > 0 ? needed : 1);
  }
  kv_bulk_copy_async<<<copyBlocks, 256, 0, stream>>>(
      (const v4f*)kc, (const v4f*)vc, (v4f*)out, nF4, units);

  // ---- Pass 2: scatter new k/v rows (32 MiB traffic) -----------------------
  const int f4row = HEAD_DIM / 4;                               // 32
  const long long total = 2LL * nkv * chunk * f4row;            // 1,048,576
  const int sblocks = (int)((total + 255) / 256);
  kv_scatter<<<sblocks, 256, 0, stream>>>(
      (const v4f*)k, (const v4f*)v, pos, (v4f*)out,
      chunk, max_ctx, nF4, nkv, f4row);
}